// ConditionalRandomField_23553600651411
// MI455X (gfx1250) — compile-verified
//
#include <hip/hip_runtime.h>
#include <hip/hip_bf16.h>

typedef float v8f __attribute__((ext_vector_type(8)));
typedef float v2f __attribute__((ext_vector_type(2)));

#define B_  512
#define L_  1024
#define T_  48
#define LT_ (L_ * T_)
#define PSTR 50   // padded LDS row stride in floats (even -> 8B-aligned b64 reads)

#define LOG2E 1.4426950408889634f
#define LN2   0.6931471805599453f

// Raw hardware transcendentals (V_EXP_F32 / V_LOG_F32 are base-2).
// Safe here: exp inputs are <= 0 (underflow->0 is correct), log inputs are
// >= e^{-O(1)} (never zero/denormal/inf).
__device__ __forceinline__ float fast_exp(float x) {
    return __builtin_amdgcn_exp2f(x * LOG2E);
}
__device__ __forceinline__ float fast_log(float x) {
    return __builtin_amdgcn_logf(x) * LN2;
}

// ---------------------------------------------------------------------------
// Gold path score: one wave per batch row -> d_ws[b]
// ---------------------------------------------------------------------------
__global__ void crf_gold_kernel(const float* __restrict__ feats,
                                const float* __restrict__ trans,
                                const int*   __restrict__ tags,
                                const int*   __restrict__ mask,
                                float*       __restrict__ gold)
{
    const int w    = (blockIdx.x * blockDim.x + threadIdx.x) >> 5; // batch row
    const int lane = threadIdx.x & 31;
    if (w >= B_) return;

    float acc = 0.0f;
    for (int l = lane; l < L_; l += 32) {
        const int   tg = tags[w * L_ + l];
        const float mk = (float)mask[w * L_ + l];
        acc += feats[(size_t)w * LT_ + (size_t)l * T_ + tg] * mk;   // emit score
        if (l >= 1) {
            const int tp = tags[w * L_ + l - 1];
            acc += trans[tp * T_ + tg] * mk;                        // trans score
        }
    }
#pragma unroll
    for (int off = 16; off; off >>= 1) acc += __shfl_xor(acc, off);
    if (lane == 0) gold[w] = acc;
}

// ---------------------------------------------------------------------------
// Forward algorithm: 3 waves per 16-row batch tile (one wave per 16-col tile).
// Per step:  alpha -> rowmax (shuffle + LDS combine) -> p = exp(alpha-m) in LDS
//            -> 12x V_WMMA_F32_16X16X4_F32 (p @ exp(trans)) -> m + log + emit.
// exp(trans) B-fragments live in VGPRs for the whole kernel.
// ---------------------------------------------------------------------------
__launch_bounds__(96, 4)
__global__ void crf_forward_kernel(const float* __restrict__ feats,
                                   const float* __restrict__ trans,
                                   const int*   __restrict__ mask,
                                   const float* __restrict__ gold,
                                   float*       __restrict__ out)
{
    __shared__ float pbuf[16 * PSTR];
    __shared__ float rowmax[3][16];
    __shared__ float rowsum[3][16];

    const int tid  = threadIdx.x;
    const int wid  = tid >> 5;          // N-tile 0..2
    const int lane = tid & 31;
    const int h    = lane >> 4;         // lane half
    const int n16  = lane & 15;
    const int rb   = blockIdx.x * 16;   // batch row base
    const int col  = wid * 16 + n16;    // tag column owned in B/C layout

    // ---- B fragments: E = exp(trans[k][col]); K = 4*kk + 2*h + v ----------
    v2f e[12];
#pragma unroll
    for (int kk = 0; kk < 12; ++kk) {
        const int k0 = kk * 4 + 2 * h;
        e[kk].x = fast_exp(trans[(k0 + 0) * T_ + col]);
        e[kk].y = fast_exp(trans[(k0 + 1) * T_ + col]);
    }

    // ---- alpha0 = feats[:, 0, :] in C layout: reg r <-> row rb + r + 8h ----
    float c[8];
#pragma unroll
    for (int r = 0; r < 8; ++r)
        c[r] = feats[(size_t)(rb + r + 8 * h) * LT_ + col];

    // ---- prefetch emissions/mask for step 1 --------------------------------
    float ecur[8]; int mcur[8];
#pragma unroll
    for (int r = 0; r < 8; ++r) {
        const int row = rb + r + 8 * h;
        ecur[r] = feats[(size_t)row * LT_ + (size_t)1 * T_ + col];
        mcur[r] = mask[row * L_ + 1];
    }

    for (int l = 1; l < L_; ++l) {
        // software pipeline: fetch step l+1 while computing step l
        float enxt[8]; int mnxt[8];
        if (l + 1 < L_) {
#pragma unroll
            for (int r = 0; r < 8; ++r) {
                const int row = rb + r + 8 * h;
                enxt[r] = feats[(size_t)row * LT_ + (size_t)(l + 1) * T_ + col];
                mnxt[r] = mask[row * L_ + l + 1];
            }
        } else {
#pragma unroll
            for (int r = 0; r < 8; ++r) { enxt[r] = 0.0f; mnxt[r] = 0; }
        }

        // per-row max of this wave's 16x16 tile (xor<16 stays in half)
        float tmax[8];
#pragma unroll
        for (int r = 0; r < 8; ++r) {
            float v = c[r];
            v = fmaxf(v, __shfl_xor(v, 1));
            v = fmaxf(v, __shfl_xor(v, 2));
            v = fmaxf(v, __shfl_xor(v, 4));
            v = fmaxf(v, __shfl_xor(v, 8));
            tmax[r] = v;
        }
        if (n16 == 0) {
#pragma unroll
            for (int r = 0; r < 8; ++r) rowmax[wid][r + 8 * h] = tmax[r];
        }
        __syncthreads();   // also fences prev-step pbuf reads vs new writes

        // combined row max; p = exp(alpha - m) staged into LDS
        float m[8];
#pragma unroll
        for (int r = 0; r < 8; ++r) {
            const int row = r + 8 * h;
            const float mm = fmaxf(rowmax[0][row],
                            fmaxf(rowmax[1][row], rowmax[2][row]));
            m[r] = mm;
            pbuf[row * PSTR + col] = fast_exp(c[r] - mm);
        }
        __syncthreads();

        // acc = p @ E for this wave's 16 columns, 12 K-steps of FP32 WMMA
        v8f acc = {};
#pragma unroll
        for (int kk = 0; kk < 12; ++kk) {
            v2f a = *(const v2f*)&pbuf[n16 * PSTR + kk * 4 + 2 * h];
            acc = __builtin_amdgcn_wmma_f32_16x16x4_f32(
                      false, a, false, e[kk], (short)0, acc, false, false);
        }

        // branchless alpha update with mask select
#pragma unroll
        for (int r = 0; r < 8; ++r) {
            const float nv = m[r] + fast_log(acc[r]) + ecur[r];
            c[r] = (mcur[r] > 0) ? nv : c[r];
        }
#pragma unroll
        for (int r = 0; r < 8; ++r) { ecur[r] = enxt[r]; mcur[r] = mnxt[r]; }
    }

    // ---- logZ = m + log(sum_t exp(alpha - m)), combined over 3 tiles -------
    float tmax[8];
#pragma unroll
    for (int r = 0; r < 8; ++r) {
        float v = c[r];
        v = fmaxf(v, __shfl_xor(v, 1));
        v = fmaxf(v, __shfl_xor(v, 2));
        v = fmaxf(v, __shfl_xor(v, 4));
        v = fmaxf(v, __shfl_xor(v, 8));
        tmax[r] = v;
    }
    if (n16 == 0) {
#pragma unroll
        for (int r = 0; r < 8; ++r) rowmax[wid][r + 8 * h] = tmax[r];
    }
    __syncthreads();

    float m[8];
#pragma unroll
    for (int r = 0; r < 8; ++r) {
        const int row = r + 8 * h;
        m[r] = fmaxf(rowmax[0][row], fmaxf(rowmax[1][row], rowmax[2][row]));
        float s = fast_exp(c[r] - m[r]);
        s += __shfl_xor(s, 1);
        s += __shfl_xor(s, 2);
        s += __shfl_xor(s, 4);
        s += __shfl_xor(s, 8);
        tmax[r] = s;    // reuse as partial sums
    }
    if (n16 == 0) {
#pragma unroll
        for (int r = 0; r < 8; ++r) rowsum[wid][r + 8 * h] = tmax[r];
    }
    __syncthreads();

    if (wid == 0 && n16 == 0) {
#pragma unroll
        for (int r = 0; r < 8; ++r) {
            const int row = r + 8 * h;
            const float tot = rowsum[0][row] + rowsum[1][row] + rowsum[2][row];
            out[rb + row] = m[r] + fast_log(tot) - gold[rb + row];
        }
    }
}

// ---------------------------------------------------------------------------
extern "C" void kernel_launch(void* const* d_in, const int* in_sizes, int n_in,
                              void* d_out, int out_size, void* d_ws, size_t ws_size,
                              hipStream_t stream) {
    const float* feats = (const float*)d_in[0];
    const float* trans = (const float*)d_in[1];
    const int*   tags  = (const int*)  d_in[2];
    const int*   mask  = (const int*)  d_in[3];
    float*       out   = (float*)d_out;
    float*       gold  = (float*)d_ws;   // B_ floats of scratch

    // 1) gold path scores (one wave per sequence)
    crf_gold_kernel<<<(B_ * 32) / 256, 256, 0, stream>>>(feats, trans, tags, mask, gold);

    // 2) forward algorithm + final NLL (3 waves per 16-row batch tile)
    crf_forward_kernel<<<B_ / 16, 96, 0, stream>>>(feats, trans, mask, gold, out);
}